// DCTBlur_24696061952732
// MI455X (gfx1250) — compile-verified
//
#include <hip/hip_runtime.h>
#include <math.h>

// ---------------------------------------------------------------------------
// DCT blur: out = C^T * (G_b ∘ (C * X * C^T)) * C  per (batch, channel) image
// N = 256. Four batched 256^3 GEMMs on V_WMMA_F32_16X16X4_F32 (fp32 WMMA).
// Staging uses CDNA5 async mem->LDS (GLOBAL_LOAD_ASYNC_TO_LDS_*, ASYNCcnt).
// Wave tile 32x64 (8 f32 accumulators), block tile 64x256, 8 waves/block.
// ---------------------------------------------------------------------------

typedef __attribute__((ext_vector_type(2))) float v2f;
typedef __attribute__((ext_vector_type(8))) float v8f;

#define DN   256          // matrix dimension
#define KC   64           // K chunk staged in LDS
#define BMT  64           // block tile rows
#define BNT  256          // block tile cols (full N)
#define LDA  68           // LDS stride for A  [m][k]  (floats)
#define LDB  68           // LDS stride for B  [n][k]  (floats)
#define SMEM_FLOATS (BMT * LDA + BNT * LDB)

#define PI_OVER_N 0.012271846303085129f       // pi/256
#define LOG_MIN_BLUR (-0.6931471805599453f)   // ln(0.5)
#define LOG_MAX_BLUR ( 2.995732273553991f)    // ln(20)

// ---- CDNA5 async mem->LDS helpers (ASYNCcnt-tracked, per-lane LDS dest) ----
__device__ __forceinline__ void async_ld_b128(unsigned lds_byte, const float* g) {
    asm volatile("global_load_async_to_lds_b128 %0, %1, off"
                 :: "v"(lds_byte), "v"((unsigned long long)g) : "memory");
}
__device__ __forceinline__ void async_ld_b32(unsigned lds_byte, const float* g) {
    asm volatile("global_load_async_to_lds_b32 %0, %1, off"
                 :: "v"(lds_byte), "v"((unsigned long long)g) : "memory");
}
__device__ __forceinline__ void wait_async0() {
    asm volatile("s_wait_asynccnt 0x0" ::: "memory");
}

// Build orthonormal DCT-II matrix C[k][h] = cos(pi*(h+0.5)*k/N) * scale(k)
__global__ void build_dct_kernel(float* __restrict__ Cmat) {
    int idx = blockIdx.x * blockDim.x + threadIdx.x;   // 65536 total
    int k = idx >> 8;
    int h = idx & 255;
    double ang = M_PI * ((double)h + 0.5) * (double)k / 256.0;
    double s   = (k == 0) ? 0.0625 : 0.08838834764831845;  // sqrt(1/256), sqrt(2/256)
    Cmat[idx] = (float)(cos(ang) * s);
}

// MODE 1: U = X * C^T          A = x    (batched, [m][k])  B = Cmat (shared, already [n][k])
// MODE 2: D = fade ∘ (C * U)   A = Cmat (shared, [m][k])   B = U    (batched, [k][n] -> transpose)
// MODE 3: V = D * C            A = D    (batched, [m][k])  B = Cmat (shared, [k][n] -> transpose)
// MODE 4: out = C^T * V        A = Cmat (shared, [k][m] -> transpose)  B = V (batched, transpose)
template <int MODE>
__global__ __launch_bounds__(256)
void dct_gemm_kernel(const float* __restrict__ A, const float* __restrict__ B,
                     float* __restrict__ Out, const float* __restrict__ t) {
    extern __shared__ float smem[];
    float* As = smem;               // [BMT][LDA]
    float* Bs = smem + BMT * LDA;   // [BNT][LDB]
    const unsigned asBase = (unsigned)(size_t)(void*)As;   // flat addr[31:0] == LDS offset
    const unsigned bsBase = (unsigned)(size_t)(void*)Bs;

    const int tid    = threadIdx.x;
    const int img    = blockIdx.y;                 // b*3 + c  (384 images)
    const int tileM0 = (int)blockIdx.x * BMT;      // 4 tiles along M, N fully covered

    const size_t ioff = (size_t)img * (DN * DN);
    const float* Ab = A + ((MODE == 2 || MODE == 4) ? 0 : ioff);
    const float* Bb = B + ((MODE == 1 || MODE == 3) ? 0 : ioff);
    float*       Ob = Out + ioff;

    float tau = 0.0f;
    if (MODE == 2) {
        float tt  = t[img / 3];
        float sig = __expf(LOG_MIN_BLUR * (1.0f - tt) + LOG_MAX_BLUR * tt);
        tau = 0.5f * sig * sig;
    }

    const int wid  = tid >> 5;
    const int lane = tid & 31;
    const int lm   = lane & 15;        // N column / M row-in-half
    const int hi   = lane >> 4;        // K-pair select / M-half select
    const int wm   = (wid >> 2) * 32;  // wave M offset within block tile (0,32)
    const int wn   = (wid & 3) * 64;   // wave N offset within block tile (0..192)

    v8f acc[2][4];
    #pragma unroll
    for (int i = 0; i < 2; ++i)
        #pragma unroll
        for (int j = 0; j < 4; ++j) acc[i][j] = (v8f){};

    #pragma unroll 1
    for (int kt = 0; kt < DN; kt += KC) {
        // ---- stage A into LDS as [m][k] via async mem->LDS ----
        if (MODE == 4) {
            // As[m][k] = C[(kt+k)*N + tileM0+m]  (transposed scatter, b32 async)
            #pragma unroll
            for (int i = 0; i < 16; ++i) {
                int idx = tid + i * 256;           // 64x64 elements
                int m = idx & 63, k = idx >> 6;
                async_ld_b32(asBase + (unsigned)(m * LDA + k) * 4,
                             &Ab[(size_t)(kt + k) * DN + tileM0 + m]);
            }
        } else {
            #pragma unroll
            for (int i = 0; i < 4; ++i) {
                int idx4 = tid + i * 256;          // 1024 float4
                int m = idx4 >> 4, k4 = idx4 & 15;
                async_ld_b128(asBase + (unsigned)(m * LDA + k4 * 4) * 4,
                              &Ab[(size_t)(tileM0 + m) * DN + kt + k4 * 4]);
            }
        }
        // ---- stage B into LDS as [n][k] via async mem->LDS ----
        if (MODE == 1) {
            // source C row-major read at [n][k] is already the layout we want
            #pragma unroll
            for (int i = 0; i < 16; ++i) {
                int idx4 = tid + i * 256;          // 4096 float4
                int n = idx4 >> 4, k4 = idx4 & 15;
                async_ld_b128(bsBase + (unsigned)(n * LDB + k4 * 4) * 4,
                              &Bb[(size_t)n * DN + kt + k4 * 4]);
            }
        } else {
            // source is [k][n] row-major -> transposed scatter (b32 async)
            #pragma unroll
            for (int k = 0; k < KC; ++k) {
                async_ld_b32(bsBase + (unsigned)(tid * LDB + k) * 4,
                             &Bb[(size_t)(kt + k) * DN + tid]);
            }
        }
        wait_async0();          // own asyncs complete (writes visible in LDS)
        __syncthreads();        // everyone's staging visible

        // ---- compute: 16 k-steps of 4; 8 WMMAs per step per wave ----
        #pragma unroll
        for (int kk = 0; kk < KC; kk += 4) {
            // f32 A 16x4 layout: lanes 0-15 hold K-pair {kk,kk+1}, lanes 16-31 {kk+2,kk+3}
            v2f a0 = *(const v2f*)&As[(wm + lm) * LDA + kk + hi * 2];
            v2f a1 = *(const v2f*)&As[(wm + 16 + lm) * LDA + kk + hi * 2];
            v2f b0 = *(const v2f*)&Bs[(wn + lm) * LDB + kk + hi * 2];
            v2f b1 = *(const v2f*)&Bs[(wn + 16 + lm) * LDB + kk + hi * 2];
            v2f b2 = *(const v2f*)&Bs[(wn + 32 + lm) * LDB + kk + hi * 2];
            v2f b3 = *(const v2f*)&Bs[(wn + 48 + lm) * LDB + kk + hi * 2];
            acc[0][0] = __builtin_amdgcn_wmma_f32_16x16x4_f32(false, a0, false, b0, (short)0, acc[0][0], false, false);
            acc[0][1] = __builtin_amdgcn_wmma_f32_16x16x4_f32(false, a0, false, b1, (short)0, acc[0][1], false, false);
            acc[0][2] = __builtin_amdgcn_wmma_f32_16x16x4_f32(false, a0, false, b2, (short)0, acc[0][2], false, false);
            acc[0][3] = __builtin_amdgcn_wmma_f32_16x16x4_f32(false, a0, false, b3, (short)0, acc[0][3], false, false);
            acc[1][0] = __builtin_amdgcn_wmma_f32_16x16x4_f32(false, a1, false, b0, (short)0, acc[1][0], false, false);
            acc[1][1] = __builtin_amdgcn_wmma_f32_16x16x4_f32(false, a1, false, b1, (short)0, acc[1][1], false, false);
            acc[1][2] = __builtin_amdgcn_wmma_f32_16x16x4_f32(false, a1, false, b2, (short)0, acc[1][2], false, false);
            acc[1][3] = __builtin_amdgcn_wmma_f32_16x16x4_f32(false, a1, false, b3, (short)0, acc[1][3], false, false);
        }
        __syncthreads();        // all reads done before next chunk overwrites LDS
    }

    // ---- epilogue: C/D layout -> global (VGPR r: M = r + hi*8, N = lm) ----
    #pragma unroll
    for (int i = 0; i < 2; ++i) {
        const int rowBase = tileM0 + wm + i * 16 + hi * 8;
        #pragma unroll
        for (int j = 0; j < 4; ++j) {
            const int col = wn + j * 16 + lm;
            float fc2 = 0.0f;
            if (MODE == 2) { float fc = PI_OVER_N * (float)col; fc2 = fc * fc; }
            #pragma unroll
            for (int r = 0; r < 8; ++r) {
                int   row = rowBase + r;
                float v = acc[i][j][r];
                if (MODE == 2) {
                    float fr = PI_OVER_N * (float)row;
                    float f  = __expf(-(fr * fr + fc2) * tau);
                    if (f < 0.01f) f = 0.0f;
                    v *= f * 0.999f + 0.001f;
                }
                Ob[(size_t)row * DN + col] = v;
            }
        }
    }
}

extern "C" void kernel_launch(void* const* d_in, const int* in_sizes, int n_in,
                              void* d_out, int out_size, void* d_ws, size_t ws_size,
                              hipStream_t stream) {
    const float* x = (const float*)d_in[0];   // (128,3,256,256)
    const float* t = (const float*)d_in[1];   // (128,)
    float* out = (float*)d_out;

    // Workspace: [0, 65536)            DCT matrix C (fp32)
    //            [65536, +384*65536)   ping buffer (100.7 MB)
    // d_out doubles as the second ping-pong buffer.
    float* Cmat = (float*)d_ws;
    float* buf0 = Cmat + 65536;

    build_dct_kernel<<<dim3(256), dim3(256), 0, stream>>>(Cmat);

    const size_t smem = SMEM_FLOATS * sizeof(float);   // 87,040 B (<320 KB/WG)
    dim3 grid(4, 384), block(256);
    dct_gemm_kernel<1><<<grid, block, smem, stream>>>(x,    Cmat, buf0, t); // U = X C^T
    dct_gemm_kernel<2><<<grid, block, smem, stream>>>(Cmat, buf0, out,  t); // D = fade∘(C U)
    dct_gemm_kernel<3><<<grid, block, smem, stream>>>(out,  Cmat, buf0, t); // V = D C
    dct_gemm_kernel<4><<<grid, block, smem, stream>>>(Cmat, buf0, out,  t); // out = C^T V
}